// SplatControl_12154757448194
// MI455X (gfx1250) — compile-verified
//
#include <hip/hip_runtime.h>
#include <hip/hip_bf16.h>

typedef __attribute__((ext_vector_type(2))) float v2f;
typedef __attribute__((ext_vector_type(8))) float v8f;

#define NBATCH   2
#define NSPLAT   2048
#define NPTS     8192
#define PTS_PER_BLOCK 64                          // 4 waves x 16 points
#define BLOCKS_PER_BATCH (NPTS / PTS_PER_BLOCK)   // 128
#define NSPLIT   8                                // splat-dimension split
#define CHUNK    (NSPLAT / NSPLIT)                // 256 splats per chunk
#define TILES_PER_CHUNK (CHUNK / 16)              // 16 WMMA tiles per chunk

// -0.5 * log2(e): folds exp(-0.5*x) into exp2(dot)
#define CC (-0.72134752044448170368f)

// ---------------------------------------------------------------------------
// Kernel 1: per-splat preprocessing.
// Sigma = R diag(s^2) R^T  =>  Sigma_inv = R diag(1/s^2) R^T  (R orthonormal).
// Emit 12-wide feature vector F (scaled by CC) and opacity*[r,g,b,1].
// ---------------------------------------------------------------------------
__global__ void splat_pre_kernel(const float* __restrict__ splats,
                                 float* __restrict__ sf,
                                 float4* __restrict__ colop) {
    int i = blockIdx.x * blockDim.x + threadIdx.x;
    if (i >= NBATCH * NSPLAT) return;
    const float* s = splats + (size_t)i * 14;
    float px = s[0], py = s[1], pz = s[2];
    float cr = s[3], cg = s[4], cb = s[5];
    float sx = s[6], sy = s[7], sz = s[8];
    float qw = s[9], qx = s[10], qy = s[11], qz = s[12];
    float op = s[13];

    float wx = 2.f*qw*qx, wy = 2.f*qw*qy, wz = 2.f*qw*qz;
    float xx = 2.f*qx*qx, xy = 2.f*qx*qy, xz = 2.f*qx*qz;
    float yy = 2.f*qy*qy, yz = 2.f*qy*qz, zz = 2.f*qz*qz;
    float r00 = 1.f-yy-zz, r01 = xy-wz,     r02 = xz+wy;
    float r10 = xy+wz,     r11 = 1.f-xx-zz, r12 = yz-wx;
    float r20 = xz-wy,     r21 = yz+wx,     r22 = 1.f-xx-yy;

    float w0 = 1.f/(sx*sx), w1 = 1.f/(sy*sy), w2 = 1.f/(sz*sz);

    float a00 = r00*r00*w0 + r01*r01*w1 + r02*r02*w2;
    float a01 = r00*r10*w0 + r01*r11*w1 + r02*r12*w2;
    float a02 = r00*r20*w0 + r01*r21*w1 + r02*r22*w2;
    float a11 = r10*r10*w0 + r11*r11*w1 + r12*r12*w2;
    float a12 = r10*r20*w0 + r11*r21*w1 + r12*r22*w2;
    float a22 = r20*r20*w0 + r21*r21*w1 + r22*r22*w2;

    float lx = a00*px + a01*py + a02*pz;
    float ly = a01*px + a11*py + a12*pz;
    float lz = a02*px + a12*py + a22*pz;
    float sv = px*lx + py*ly + pz*lz;

    float* f = sf + (size_t)i * 12;
    f[0]  = CC * a00;        f[1]  = CC * a11;        f[2]  = CC * a22;
    f[3]  = CC * 2.f * a01;  f[4]  = CC * 2.f * a02;  f[5]  = CC * 2.f * a12;
    f[6]  = CC * -2.f * lx;  f[7]  = CC * -2.f * ly;  f[8]  = CC * -2.f * lz;
    f[9]  = CC * sv;         f[10] = 0.f;             f[11] = 0.f;

    float4 co; co.x = op*cr; co.y = op*cg; co.z = op*cb; co.w = op;
    colop[i] = co;
}

// ---------------------------------------------------------------------------
// Kernel 2: main pairwise kernel (one splat chunk per block).
// Each wave owns 16 points; iterates over 16 splat tiles of 16.
// mahal^T tile (splats x points) via 3x V_WMMA_F32_16X16X4_F32 (K = 12),
// density via v_exp_f32, opacity-weighted color accumulated per lane.
// Writes a per-chunk partial (num.xyz, den) per point -- no atomics.
// ---------------------------------------------------------------------------
__global__ void __launch_bounds__(128)
splat_main_kernel(const float* __restrict__ coords,
                  const float* __restrict__ sf,
                  const float4* __restrict__ colop,
                  float4* __restrict__ part) {
    __shared__ float4 colop_s[CHUNK];   // 4 KB static LDS

    const int tid   = threadIdx.x;
    const int lane  = tid & 31;
    const int wave  = tid >> 5;
    const int chunk = blockIdx.x / (NBATCH * BLOCKS_PER_BATCH);
    const int rem   = blockIdx.x % (NBATCH * BLOCKS_PER_BATCH);
    const int b     = rem / BLOCKS_PER_BATCH;
    const int mt    = rem % BLOCKS_PER_BATCH;
    const int nbase = chunk * CHUNK;    // first splat of this chunk

    // Stage this chunk's colop table into LDS (broadcast-heavy inner loop).
    const float4* cog = colop + (size_t)b * NSPLAT + nbase;
    for (int i = tid; i < CHUNK; i += 128) colop_s[i] = cog[i];
    __syncthreads();

    const bool hi    = (lane >= 16);
    const int  col   = lane & 15;
    const int  point = mt * PTS_PER_BLOCK + wave * 16 + col;

    // Point features (B fragment: 12 x 16 fp32; lanes 0-15 hold K={0,1},
    // lanes 16-31 hold K={2,3} within each 16x16x4 step).
    const float* cp = coords + ((size_t)b * NPTS + point) * 3;
    float cx = cp[0], cy = cp[1], cz = cp[2];
    float f0 = cx*cx, f1 = cy*cy, f2 = cz*cz;
    float f3 = cx*cy, f4 = cx*cz, f5 = cy*cz;
    float f6 = cx,    f7 = cy,    f8 = cz,   f9 = 1.f;

    v2f bf0, bf1, bf2;
    bf0[0] = hi ? f2  : f0; bf0[1] = hi ? f3  : f1;
    bf1[0] = hi ? f6  : f4; bf1[1] = hi ? f7  : f5;
    bf2[0] = hi ? 0.f : f8; bf2[1] = hi ? 0.f : f9;

    const float* sfb  = sf + ((size_t)b * NSPLAT + nbase) * 12;
    const int    koff = hi ? 2 : 0;

    float acc0 = 0.f, acc1 = 0.f, acc2 = 0.f, acc3 = 0.f;

    #pragma unroll 2
    for (int nt = 0; nt < TILES_PER_CHUNK; ++nt) {
        // A fragment: 16 splats x 12 features, same lane-half K split.
        const float* fa = sfb + (size_t)(nt * 16 + col) * 12 + koff;
        v2f af0, af1, af2;
        af0[0] = fa[0]; af0[1] = fa[1];
        af1[0] = fa[4]; af1[1] = fa[5];
        af2[0] = fa[8]; af2[1] = fa[9];

        v8f c = {};
        c = __builtin_amdgcn_wmma_f32_16x16x4_f32(false, af0, false, bf0,
                                                  (short)0, c, false, false);
        c = __builtin_amdgcn_wmma_f32_16x16x4_f32(false, af1, false, bf1,
                                                  (short)0, c, false, false);
        c = __builtin_amdgcn_wmma_f32_16x16x4_f32(false, af2, false, bf2,
                                                  (short)0, c, false, false);

        // C layout: lane holds rows M = r + (hi?8:0), one point column.
        const int rbase = nt * 16 + (hi ? 8 : 0);
        #pragma unroll
        for (int r = 0; r < 8; ++r) {
            float d = __builtin_amdgcn_exp2f(c[r]);   // v_exp_f32
            float4 co = colop_s[rbase + r];           // ds_load_b128 broadcast
            acc0 += d * co.x;
            acc1 += d * co.y;
            acc2 += d * co.z;
            acc3 += d * co.w;
        }
    }

    // Combine the two lane-halves (splat rows 0-7 + 8-15 of every tile).
    acc0 += __shfl_xor(acc0, 16);
    acc1 += __shfl_xor(acc1, 16);
    acc2 += __shfl_xor(acc2, 16);
    acc3 += __shfl_xor(acc3, 16);

    if (!hi) {
        float4 p; p.x = acc0; p.y = acc1; p.z = acc2; p.w = acc3;
        part[(size_t)chunk * (NBATCH * NPTS) + (size_t)b * NPTS + point] = p;
    }
}

// ---------------------------------------------------------------------------
// Kernel 3: finalize -- sum NSPLIT partials per point, divide by density sum.
// ---------------------------------------------------------------------------
__global__ void splat_finalize_kernel(const float4* __restrict__ part,
                                      float* __restrict__ out) {
    int gp = blockIdx.x * blockDim.x + threadIdx.x;   // b*NPTS + point
    if (gp >= NBATCH * NPTS) return;
    float n0 = 0.f, n1 = 0.f, n2 = 0.f, dn = 0.f;
    #pragma unroll
    for (int s = 0; s < NSPLIT; ++s) {
        float4 p = part[(size_t)s * (NBATCH * NPTS) + gp];
        n0 += p.x; n1 += p.y; n2 += p.z; dn += p.w;
    }
    float inv = 1.f / (dn + 1e-6f);
    float* o = out + (size_t)gp * 3;
    o[0] = n0 * inv;
    o[1] = n1 * inv;
    o[2] = n2 * inv;
}

// ---------------------------------------------------------------------------
extern "C" void kernel_launch(void* const* d_in, const int* in_sizes, int n_in,
                              void* d_out, int out_size, void* d_ws, size_t ws_size,
                              hipStream_t stream) {
    const float* splats = (const float*)d_in[0];   // (2, 2048, 14) f32
    const float* coords = (const float*)d_in[1];   // (2, 8192, 3)  f32

    float*  sf    = (float*)d_ws;                          // 2*2048*12 f32
    float4* colop = (float4*)(sf + NBATCH * NSPLAT * 12);  // 2*2048 float4
    float4* part  = colop + NBATCH * NSPLAT;               // 8*2*8192 float4 (2 MB)

    splat_pre_kernel<<<(NBATCH * NSPLAT + 255) / 256, 256, 0, stream>>>(
        splats, sf, colop);

    splat_main_kernel<<<NSPLIT * NBATCH * BLOCKS_PER_BATCH, 128, 0, stream>>>(
        coords, sf, colop, part);

    splat_finalize_kernel<<<(NBATCH * NPTS + 255) / 256, 256, 0, stream>>>(
        part, (float*)d_out);
}